// QuantumQuanvolutionHybrid_65481071398168
// MI455X (gfx1250) — compile-verified
//
#include <hip/hip_runtime.h>
#include <math.h>

typedef __attribute__((ext_vector_type(2))) float v2f;
typedef __attribute__((ext_vector_type(8))) float v8f;

#define BATCH   4096
#define NPIX    784          // 28*28, also feature count (196 patches * 4)
#define NCLS    10
#define NPATCH  (BATCH * 196)
#define WSTRIDE 788          // 784 + 4 pad: n*788 mod 64 banks -> 16 distinct banks

// ---------------------------------------------------------------------------
// Kernel 1: quantum patch filter. One thread = one 2x2 patch. The 16-amplitude
// real state vector lives entirely in registers; every gate loop is unrolled
// so strides (8>>q) constant-fold and the whole circuit becomes straight-line
// FMA code. Writes feats[b*784 + patch*4 + q] matching the reference layout.
// ---------------------------------------------------------------------------
__device__ __forceinline__ void apply_ry(float* s, float c, float sn, int q) {
    const int stride = 8 >> q;           // qubit 0 is the MSB of the amp index
#pragma unroll
    for (int idx = 0; idx < 16; ++idx) {
        if ((idx & stride) == 0) {
            float a0 = s[idx], a1 = s[idx + stride];
            s[idx]          = c * a0 - sn * a1;
            s[idx + stride] = sn * a0 + c * a1;
        }
    }
}

__device__ __forceinline__ void apply_cnot(float* s, int cq, int tq) {
    const int cs = 8 >> cq, ts = 8 >> tq;
#pragma unroll
    for (int idx = 0; idx < 16; ++idx) {
        if ((idx & cs) != 0 && (idx & ts) == 0) {   // control=1, target=0
            float t = s[idx];
            s[idx] = s[idx | ts];
            s[idx | ts] = t;
        }
    }
}

__global__ __launch_bounds__(256)
void quanv_kernel(const float* __restrict__ x,
                  const float* __restrict__ vp,
                  float* __restrict__ feats) {
    int p = blockIdx.x * blockDim.x + threadIdx.x;   // NPATCH divisible by 256
    if (p >= NPATCH) return;
    int bimg = p / 196;
    int pi   = p - bimg * 196;
    int i = pi / 14, j = pi - i * 14;

    const float* base = x + bimg * NPIX + (2 * i) * 28 + 2 * j;
    float d[4] = { base[0], base[1], base[28], base[29] };   // row-major 2x2

    float cd[4], sd[4], cp[4], sp[4];
#pragma unroll
    for (int q = 0; q < 4; ++q) sincosf(0.5f * d[q],  &sd[q], &cd[q]);
#pragma unroll
    for (int k = 0; k < 4; ++k) sincosf(0.5f * vp[k], &sp[k], &cp[k]);

    float s[16];
#pragma unroll
    for (int t = 0; t < 16; ++t) s[t] = 0.0f;
    s[0] = 1.0f;                                             // |0000>

#pragma unroll
    for (int layer = 0; layer < 2; ++layer) {
#pragma unroll
        for (int q = 0; q < 4; ++q) apply_ry(s, cd[q], sd[q], q);
#pragma unroll
        for (int k = 0; k < 4; ++k) {
            apply_ry(s, cp[k], sp[k], k);
            apply_cnot(s, k, (k + 1) & 3);
        }
    }

    float* fout = feats + bimg * NPIX + pi * 4;
#pragma unroll
    for (int q = 0; q < 4; ++q) {
        const int bit = 8 >> q;
        float e = 0.0f;
#pragma unroll
        for (int idx = 0; idx < 16; ++idx) {
            float pr = s[idx] * s[idx];
            e += (idx & bit) ? -pr : pr;                     // <Z_q>
        }
        fout[q] = e;
    }
}

// ---------------------------------------------------------------------------
// Kernel 2: logits = feats @ W^T + b via V_WMMA_F32_16X16X4_F32, then fused
// log_softmax. 64 blocks x 128 threads (4 waves); each wave owns one 16-image
// tile (256 tiles total, exact cover -> EXEC all-1s through the WMMA loop).
// W is zero-padded to 16 rows in LDS (stride 788 -> bank-conflict-free),
// so A and B fragments are unconditional aligned v2f loads.
//   A frag (16x4): lane L: m=L%16, VGPR0/1 = K = 2*(L/16) + {0,1}
//   B frag (4x16): lane L: n=L%16, VGPR0/1 = K = 2*(L/16) + {0,1}
//   C/D   (16x16): lane L: n=L%16, acc[v] = row m = v + 8*(L/16)
// ---------------------------------------------------------------------------
__global__ __launch_bounds__(128)
void gemm_lsm_kernel(const float* __restrict__ feats,
                     const float* __restrict__ W,
                     const float* __restrict__ bias,
                     float* __restrict__ out) {
    __shared__ float Wl[16 * WSTRIDE];                       // 50,432 B

    int tid = threadIdx.x;
    for (int idx = tid; idx < 16 * NPIX; idx += 128) {
        int row = idx / NPIX, col = idx - row * NPIX;
        Wl[row * WSTRIDE + col] = (row < NCLS) ? W[row * NPIX + col] : 0.0f;
    }
    __syncthreads();

    const int wave = tid >> 5;                // 0..3
    const int lane = tid & 31;
    const int tile = blockIdx.x * 4 + wave;   // 0..255
    const int m    = lane & 15;               // A row index == B col index n
    const int half = lane >> 4;

    const float* arow = feats + (size_t)(tile * 16 + m) * NPIX;
    const float* brow = Wl + m * WSTRIDE;

    v8f acc = {};
    for (int k = 0; k < NPIX; k += 4) {
        v2f a  = *(const v2f*)(arow + k + 2 * half);   // 8B-aligned (784,k even)
        v2f bb = *(const v2f*)(brow + k + 2 * half);
        acc = __builtin_amdgcn_wmma_f32_16x16x4_f32(
                  /*neg_a=*/false, a, /*neg_b=*/false, bb,
                  /*c_mod=*/(short)0, acc, /*reuse_a=*/false, /*reuse_b=*/false);
    }

    // Fused log_softmax: each acc[v] is one logit row spread over a 16-lane
    // half-wave (n = lane%16). Pad lanes n>=10 with -inf so max/sum ignore them.
    const float bv = (m < NCLS) ? bias[m] : 0.0f;
#pragma unroll
    for (int v = 0; v < 8; ++v) {
        float logit = (m < NCLS) ? (acc[v] + bv) : -__builtin_inff();
        float mx = logit;
#pragma unroll
        for (int off = 1; off < 16; off <<= 1)
            mx = fmaxf(mx, __shfl_xor(mx, off, 32));  // xor<16 stays in half
        float se = __expf(logit - mx);                // exp(-inf)=0 on pad lanes
#pragma unroll
        for (int off = 1; off < 16; off <<= 1)
            se += __shfl_xor(se, off, 32);
        float lse = mx + __logf(se);
        if (m < NCLS) {
            int row = tile * 16 + v + 8 * half;
            out[row * NCLS + m] = logit - lse;
        }
    }
}

// ---------------------------------------------------------------------------
extern "C" void kernel_launch(void* const* d_in, const int* in_sizes, int n_in,
                              void* d_out, int out_size, void* d_ws, size_t ws_size,
                              hipStream_t stream) {
    (void)in_sizes; (void)n_in; (void)out_size; (void)ws_size;
    const float* x    = (const float*)d_in[0];   // [4096, 784]
    const float* vp   = (const float*)d_in[1];   // [4]
    const float* W    = (const float*)d_in[2];   // [10, 784]
    const float* bias = (const float*)d_in[3];   // [10]
    float* out   = (float*)d_out;                // [4096, 10]
    float* feats = (float*)d_ws;                 // [4096, 784] scratch (12.85 MB)

    quanv_kernel<<<NPATCH / 256, 256, 0, stream>>>(x, vp, feats);
    gemm_lsm_kernel<<<64, 128, 0, stream>>>(feats, W, bias, out);
}